// WaveMambaBlock_86492051406979
// MI455X (gfx1250) — compile-verified
//
#include <hip/hip_runtime.h>
#include <hip/hip_bf16.h>
#include <math.h>
#include <stdint.h>

// ---------------------------------------------------------------------------
// WaveMambaBlock for MI455X (gfx1250, wave32). f32 WMMA (16x16x4) for all
// GEMM-shaped stages; 3-phase chunked selective scan (chunk decay products
// collapse to exp(A*sum_dt)) with async global->LDS staging; BN stats via
// global_atomic_add_f32.
// Shapes: B=8, C=32, T=512, F=128 -> Tp=256, Fp=64, L=16384 tok/batch,
// NTOK=131072 tokens total. di=64, ds=16, dtr=2.
// ---------------------------------------------------------------------------

typedef float v2f __attribute__((ext_vector_type(2)));
typedef float v8f __attribute__((ext_vector_type(8)));

#define LTOK 16384
#define NTOK 131072
#define CH 64           // scan chunk length
#define NCH 256         // chunks per batch (LTOK / CH)

__device__ __forceinline__ v8f wmma4(v2f a, v2f b, v8f c) {
  // D = A(16x4 f32) * B(4x16 f32) + C(16x16 f32)
  return __builtin_amdgcn_wmma_f32_16x16x4_f32(false, a, false, b, (short)0, c,
                                               false, false);
}

__device__ __forceinline__ float wredsum(float v) {
  v += __shfl_xor(v, 1);
  v += __shfl_xor(v, 2);
  v += __shfl_xor(v, 4);
  v += __shfl_xor(v, 8);
  v += __shfl_xor(v, 16);
  return v;
}

__device__ __forceinline__ void atomAddF(float* p, float v) { unsafeAtomicAdd(p, v); }
__device__ __forceinline__ float siluf(float v) { return v / (1.f + __expf(-v)); }
__device__ __forceinline__ float softplusf(float v) {
  return (v > 20.f) ? v : __logf(1.f + __expf(v));
}

// Issue one per-lane 16B async global->LDS copy (ASYNCcnt-tracked TDM-lite).
__device__ __forceinline__ void async_g2l_b128(const float* gaddr, const float* laddr) {
  unsigned lds = (unsigned)(size_t)laddr;              // low 32 bits = LDS offset
  unsigned long long ga = (unsigned long long)(size_t)gaddr;
  asm volatile("global_load_async_to_lds_b128 %0, %1, off" ::"v"(lds), "v"(ga)
               : "memory");
}
__device__ __forceinline__ void wait_async0() {
  asm volatile("s_wait_asynccnt 0x0" ::: "memory");
}

// ---------------------------------------------------------------------------
// 0) prep: zero BN stats; A = -exp(A_log); compose Wc = [dt_w@xproj[:2]; B; C]
//    (96x64) and Wo = vim_out_w @ out_w (32x64), per direction.
// ---------------------------------------------------------------------------
__global__ void prep_kernel(const float* __restrict__ Af, const float* __restrict__ Ab,
                            const float* __restrict__ dtwf, const float* __restrict__ dtwb,
                            const float* __restrict__ xpf, const float* __restrict__ xpb,
                            const float* __restrict__ vimW,
                            const float* __restrict__ owf, const float* __restrict__ owb,
                            float* __restrict__ Anegf, float* __restrict__ Anegb,
                            float* __restrict__ Wcf, float* __restrict__ Wcb,
                            float* __restrict__ Wof, float* __restrict__ Wob,
                            float* __restrict__ stats) {
  int t = threadIdx.x;
  for (int i = t; i < 384; i += 256) stats[i] = 0.f;
  for (int i = t; i < 1024; i += 256) {
    Anegf[i] = -__expf(Af[i]);
    Anegb[i] = -__expf(Ab[i]);
  }
  for (int i = t; i < 6144; i += 256) {
    int n = i >> 6, k = i & 63;
    float vf, vb;
    if (n < 64) {
      vf = dtwf[n * 2] * xpf[k] + dtwf[n * 2 + 1] * xpf[64 + k];
      vb = dtwb[n * 2] * xpb[k] + dtwb[n * 2 + 1] * xpb[64 + k];
    } else if (n < 80) {
      vf = xpf[(2 + n - 64) * 64 + k];
      vb = xpb[(2 + n - 64) * 64 + k];
    } else {
      vf = xpf[(18 + n - 80) * 64 + k];
      vb = xpb[(18 + n - 80) * 64 + k];
    }
    Wcf[i] = vf;
    Wcb[i] = vb;
  }
  for (int i = t; i < 2048; i += 256) {
    int c = i >> 6, dd = i & 63;
    float sf = 0.f, sb = 0.f;
    for (int j = 0; j < 32; ++j) {
      sf += vimW[c * 32 + j] * owf[j * 64 + dd];
      sb += vimW[c * 32 + j] * owb[j * 64 + dd];
    }
    Wof[i] = sf;
    Wob[i] = sb;
  }
}

// ---------------------------------------------------------------------------
// 1) Haar + dual heads (1,2)-conv stride(1,2) + bias; accumulate BN stats.
// ---------------------------------------------------------------------------
__global__ void head_kernel(const float* __restrict__ x,
                            const float* __restrict__ wm, const float* __restrict__ bm,
                            const float* __restrict__ wa, const float* __restrict__ ba,
                            float* __restrict__ mainp, float* __restrict__ auxp,
                            float* __restrict__ stats) {
  const float Rc = 0.70710678f;
  int idx = blockIdx.x * 256 + threadIdx.x;  // 131072 pixels
  int f = idx & 63;
  int t = (idx >> 6) & 255;
  int b = idx >> 14;
  float mAcc[32], aAcc[32];
#pragma unroll
  for (int co = 0; co < 32; ++co) { mAcc[co] = bm[co]; aAcc[co] = ba[co]; }
  for (int ci = 0; ci < 32; ++ci) {
    size_t base = ((size_t)(b * 32 + ci) * 512 + 2 * t) * 128 + 2 * f;
    float x0a = x[base], x0b = x[base + 1];
    float x1a = x[base + 128], x1b = x[base + 129];
    float lo0 = (x0a + x1a) * Rc, lo1 = (x0b + x1b) * Rc;
    float hi0 = (x1a - x0a) * Rc, hi1 = (x1b - x0b) * Rc;
#pragma unroll
    for (int co = 0; co < 32; ++co) {
      mAcc[co] = fmaf(wm[co * 64 + ci * 2], lo0, fmaf(wm[co * 64 + ci * 2 + 1], lo1, mAcc[co]));
      aAcc[co] = fmaf(wa[co * 64 + ci * 2], hi0, fmaf(wa[co * 64 + ci * 2 + 1], hi1, aAcc[co]));
    }
  }
  int l = t * 64 + f;
#pragma unroll
  for (int co = 0; co < 32; ++co) {
    mainp[((size_t)(b * 32 + co)) * LTOK + l] = mAcc[co];
    auxp[((size_t)(b * 32 + co)) * LTOK + l] = aAcc[co];
  }
  __shared__ float sm[128];  // [0..63]=sum ch, [64..127]=sumsq ch (main 0-31, aux 32-63)
  if (threadIdx.x < 128) sm[threadIdx.x] = 0.f;
  __syncthreads();
  int lane = threadIdx.x & 31;
#pragma unroll
  for (int co = 0; co < 32; ++co) {
    float s = wredsum(mAcc[co]);
    float q = wredsum(mAcc[co] * mAcc[co]);
    if (lane == 0) { atomAddF(&sm[co], s); atomAddF(&sm[64 + co], q); }
    s = wredsum(aAcc[co]);
    q = wredsum(aAcc[co] * aAcc[co]);
    if (lane == 0) { atomAddF(&sm[32 + co], s); atomAddF(&sm[96 + co], q); }
  }
  __syncthreads();
  if (threadIdx.x < 64) atomAddF(&stats[threadIdx.x], sm[threadIdx.x]);
  else if (threadIdx.x < 128) atomAddF(&stats[192 + threadIdx.x - 64], sm[threadIdx.x]);
}

// ---------------------------------------------------------------------------
// BN finalize: stats -> (scale, shift) pairs in bnp
// ---------------------------------------------------------------------------
__global__ void bn_finalize_kernel(const float* __restrict__ stats,
                                   const float* __restrict__ g, const float* __restrict__ b,
                                   float* __restrict__ bnp, int off, int C, float invN) {
  int c = threadIdx.x;
  if (c < C) {
    float mean = stats[off + c] * invN;
    float var = stats[192 + off + c] * invN - mean * mean;
    float sc = g[c] * rsqrtf(var + 1e-5f);
    bnp[2 * (off + c)] = sc;
    bnp[2 * (off + c) + 1] = b[c] - mean * sc;
  }
}

// BN + activation in place on [B,32,LTOK] planar buffers. act: 0=relu 1=silu
__global__ void bn_act_kernel(float* __restrict__ buf, const float* __restrict__ bnp,
                              int off, int act) {
  int idx = blockIdx.x * 256 + threadIdx.x;
  int c = (idx >> 14) & 31;
  float v = buf[idx] * bnp[2 * (off + c)] + bnp[2 * (off + c) + 1];
  buf[idx] = act ? siluf(v) : fmaxf(v, 0.f);
}

// ---------------------------------------------------------------------------
// 2) tokens + LayerNorm: tok[m,c]=main[b,c,l]; xn = LN(tok)*g+b
// ---------------------------------------------------------------------------
__global__ void ln_kernel(const float* __restrict__ mainp,
                          const float* __restrict__ g, const float* __restrict__ bb,
                          float* __restrict__ tok, float* __restrict__ xn) {
  int m = blockIdx.x * 256 + threadIdx.x;
  int b = m >> 14, l = m & (LTOK - 1);
  float v[32];
  float s = 0.f;
#pragma unroll
  for (int c = 0; c < 32; ++c) { v[c] = mainp[((size_t)(b * 32 + c)) * LTOK + l]; s += v[c]; }
  float mean = s * (1.f / 32.f);
  float q = 0.f;
#pragma unroll
  for (int c = 0; c < 32; ++c) { float d = v[c] - mean; q += d * d; }
  float inv = rsqrtf(q * (1.f / 32.f) + 1e-5f);
#pragma unroll
  for (int c = 0; c < 32; ++c) {
    tok[(size_t)m * 32 + c] = v[c];
    xn[(size_t)m * 32 + c] = (v[c] - mean) * inv * g[c] + bb[c];
  }
}

// ---------------------------------------------------------------------------
// 3) in_proj WMMA GEMM: [u|z] = xn(@flip) @ in_w^T   M=NTOK N=128 K=32
// ---------------------------------------------------------------------------
__global__ void gemm_inproj_kernel(const float* __restrict__ xn, const float* __restrict__ inw,
                                   float* __restrict__ u_raw, float* __restrict__ zb, int flip) {
  int m0 = blockIdx.x * 16;
  int wv = threadIdx.x >> 5;   // 8 waves -> 8 N-tiles
  int lane = threadIdx.x & 31;
  int n0 = wv * 16;
  int mh = lane & 15, kh = (lane >> 4) << 1;
  int srow;
  if (flip) {
    int l0 = m0 & (LTOK - 1);
    srow = (m0 - l0) + (LTOK - 1 - l0 - mh);
  } else {
    srow = m0 + mh;
  }
  const float* A = xn + (size_t)srow * 32 + kh;
  const float* Wp = inw + (size_t)(n0 + mh) * 32 + kh;
  v8f acc = {};
#pragma unroll
  for (int k = 0; k < 32; k += 4) {
    v2f a = {A[k], A[k + 1]};
    v2f b = {Wp[k], Wp[k + 1]};
    acc = wmma4(a, b, acc);
  }
  float* dst = (n0 < 64) ? u_raw : zb;
  int n = ((n0 < 64) ? n0 : n0 - 64) + (lane & 15);
  int mb = (lane >> 4) * 8;
#pragma unroll
  for (int v = 0; v < 8; ++v) dst[(size_t)(m0 + mb + v) * 64 + n] = acc[v];
}

// ---------------------------------------------------------------------------
// 4) causal depthwise conv1d (k=4, pad 3) + bias + SiLU
// ---------------------------------------------------------------------------
__global__ void dwconv_silu_kernel(const float* __restrict__ u_raw,
                                   const float* __restrict__ cw, const float* __restrict__ cb,
                                   float* __restrict__ u_act) {
  int idx = blockIdx.x * 256 + threadIdx.x;  // NTOK*64
  int d = idx & 63;
  int r = idx >> 6;
  int l = r & (LTOK - 1);
  float acc = cb[d];
#pragma unroll
  for (int j = 0; j < 4; ++j) {
    int ll = l - 3 + j;
    if (ll >= 0) acc += cw[d * 4 + j] * u_raw[(size_t)(r - 3 + j) * 64 + d];
  }
  u_act[(size_t)r * 64 + d] = siluf(acc);
}

// ---------------------------------------------------------------------------
// 5) x-proj WMMA GEMM: dbc2 = u_act @ Wc^T  M=NTOK N=96 K=64
//    cols 0..63 dt-lin, 64..79 B, 80..95 C
// ---------------------------------------------------------------------------
__global__ void gemm_xproj_kernel(const float* __restrict__ u_act,
                                  const float* __restrict__ Wc,
                                  float* __restrict__ dbc2) {
  int m0 = blockIdx.x * 16;
  int wv = threadIdx.x >> 5;  // 6 waves
  int lane = threadIdx.x & 31;
  int n0 = wv * 16;
  int mh = lane & 15, kh = (lane >> 4) << 1;
  const float* A = u_act + (size_t)(m0 + mh) * 64 + kh;
  const float* Wp = Wc + (size_t)(n0 + mh) * 64 + kh;
  v8f acc = {};
#pragma unroll
  for (int k = 0; k < 64; k += 4) {
    v2f a = {A[k], A[k + 1]};
    v2f b = {Wp[k], Wp[k + 1]};
    acc = wmma4(a, b, acc);
  }
  int n = n0 + (lane & 15);
  int mb = (lane >> 4) * 8;
#pragma unroll
  for (int v = 0; v < 8; ++v) dbc2[(size_t)(m0 + mb + v) * 96 + n] = acc[v];
}

// ---------------------------------------------------------------------------
// 6) chunked selective scan. Recurrence h_t = exp(dt_t*A)*h_{t-1} + dt_t*B_t*u_t
//    Chunk decay product collapses: prod_t exp(dt_t*A) = exp(A * sum_t dt_t).
//    Phase 1: per-chunk local scan (h0=0) -> h_end, sum_dt.   [parallel]
//    Phase 2: per-batch combine of chunk summaries -> h_in.    [tiny serial]
//    Phase 3: per-chunk replay from h_in, emit gated y.        [parallel]
//    Phases 1/3 stage dbc2 (24KB) and u (16KB) chunks into LDS with
//    global_load_async_to_lds_b128 (ASYNCcnt) before the hot loop.
// ---------------------------------------------------------------------------
__global__ void scan_phase1_kernel(const float* __restrict__ dbc2,
                                   const float* __restrict__ u_act,
                                   const float* __restrict__ Aneg,
                                   const float* __restrict__ dtb,
                                   float* __restrict__ hend, float* __restrict__ sdt) {
  __shared__ __align__(16) float dbuf[CH * 96];
  __shared__ __align__(16) float ubuf[CH * 64];
  int blk = blockIdx.x;  // b*NCH + ch
  int b = blk >> 8;
  int ch = blk & (NCH - 1);
  int t = threadIdx.x;
  size_t row0 = (size_t)b * LTOK + ch * CH;
  {
    const float* dsrc = dbc2 + row0 * 96;
    const float* usrc = u_act + row0 * 64;
    for (int i = t * 4; i < CH * 96; i += 1024) async_g2l_b128(dsrc + i, &dbuf[i]);
    for (int i = t * 4; i < CH * 64; i += 1024) async_g2l_b128(usrc + i, &ubuf[i]);
    wait_async0();
    __syncthreads();
  }
  int d = t >> 2;
  int sb = (t & 3) << 2;
  float4 Av = *(const float4*)(Aneg + d * 16 + sb);
  float dtbias = dtb[d];
  float h0 = 0.f, h1 = 0.f, h2 = 0.f, h3 = 0.f, sd = 0.f;
  for (int l = 0; l < CH; ++l) {
    const float* dr = dbuf + l * 96;
    float dt = softplusf(dr[d] + dtbias);
    sd += dt;
    float4 Bv = *(const float4*)(dr + 64 + sb);
    float dtu = dt * ubuf[l * 64 + d];
    h0 = __expf(dt * Av.x) * h0 + dtu * Bv.x;
    h1 = __expf(dt * Av.y) * h1 + dtu * Bv.y;
    h2 = __expf(dt * Av.z) * h2 + dtu * Bv.z;
    h3 = __expf(dt * Av.w) * h3 + dtu * Bv.w;
  }
  float4 hv;
  hv.x = h0; hv.y = h1; hv.z = h2; hv.w = h3;
  *(float4*)(hend + (size_t)blk * 1024 + d * 16 + sb) = hv;
  if ((t & 3) == 0) sdt[(size_t)blk * 64 + d] = sd;
}

__global__ void scan_phase2_kernel(const float* __restrict__ hend,
                                   const float* __restrict__ sdt,
                                   const float* __restrict__ Aneg,
                                   float* __restrict__ hin) {
  int b = blockIdx.x;
  int t = threadIdx.x;
  int d = t >> 2;
  int sb = (t & 3) << 2;
  float4 Av = *(const float4*)(Aneg + d * 16 + sb);
  float4 h;
  h.x = 0.f; h.y = 0.f; h.z = 0.f; h.w = 0.f;
  for (int ch = 0; ch < NCH; ++ch) {
    size_t blk = (size_t)b * NCH + ch;
    *(float4*)(hin + blk * 1024 + d * 16 + sb) = h;
    float sd = sdt[blk * 64 + d];
    float4 he = *(const float4*)(hend + blk * 1024 + d * 16 + sb);
    h.x = __expf(Av.x * sd) * h.x + he.x;
    h.y = __expf(Av.y * sd) * h.y + he.y;
    h.z = __expf(Av.z * sd) * h.z + he.z;
    h.w = __expf(Av.w * sd) * h.w + he.w;
  }
}

__global__ void scan_phase3_kernel(const float* __restrict__ dbc2,
                                   const float* __restrict__ u_act,
                                   const float* __restrict__ zb,
                                   const float* __restrict__ Aneg,
                                   const float* __restrict__ dtb,
                                   const float* __restrict__ Dp,
                                   const float* __restrict__ hin,
                                   float* __restrict__ yg) {
  __shared__ __align__(16) float dbuf[CH * 96];
  __shared__ __align__(16) float ubuf[CH * 64];
  int blk = blockIdx.x;
  int b = blk >> 8;
  int ch = blk & (NCH - 1);
  int t = threadIdx.x;
  size_t row0 = (size_t)b * LTOK + ch * CH;
  {
    const float* dsrc = dbc2 + row0 * 96;
    const float* usrc = u_act + row0 * 64;
    for (int i = t * 4; i < CH * 96; i += 1024) async_g2l_b128(dsrc + i, &dbuf[i]);
    for (int i = t * 4; i < CH * 64; i += 1024) async_g2l_b128(usrc + i, &ubuf[i]);
    wait_async0();
    __syncthreads();
  }
  int d = t >> 2;
  int sb = (t & 3) << 2;
  float4 Av = *(const float4*)(Aneg + d * 16 + sb);
  float dtbias = dtb[d];
  float Dd = Dp[d];
  float4 h = *(const float4*)(hin + (size_t)blk * 1024 + d * 16 + sb);
  bool lead = (t & 3) == 0;
  for (int l = 0; l < CH; ++l) {
    const float* dr = dbuf + l * 96;
    float dt = softplusf(dr[d] + dtbias);
    float4 Bv = *(const float4*)(dr + 64 + sb);
    float4 Cv = *(const float4*)(dr + 80 + sb);
    float ud = ubuf[l * 64 + d];
    float dtu = dt * ud;
    h.x = __expf(dt * Av.x) * h.x + dtu * Bv.x;
    h.y = __expf(dt * Av.y) * h.y + dtu * Bv.y;
    h.z = __expf(dt * Av.z) * h.z + dtu * Bv.z;
    h.w = __expf(dt * Av.w) * h.w + dtu * Bv.w;
    float ys = h.x * Cv.x + h.y * Cv.y + h.z * Cv.z + h.w * Cv.w;
    ys += __shfl_xor(ys, 1);  // reduce across the 4-lane s-group (wave32)
    ys += __shfl_xor(ys, 2);
    if (lead) {
      size_t row = row0 + l;
      float zd = zb[row * 64 + d];
      yg[row * 64 + d] = (ys + ud * Dd) * siluf(zd);
    }
  }
}

// ---------------------------------------------------------------------------
// 7) out-proj WMMA GEMM: vim (+)= yg @ Wo^T ; mode0: init + vim_out_b + tok,
//    mode1: flipped accumulate. M=NTOK N=32 K=64
// ---------------------------------------------------------------------------
__global__ void gemm_out_kernel(const float* __restrict__ yg, const float* __restrict__ Wo,
                                const float* __restrict__ vb, const float* __restrict__ tok,
                                float* __restrict__ vim, int mode) {
  int m0 = blockIdx.x * 16;
  int wv = threadIdx.x >> 5;  // 2 waves
  int lane = threadIdx.x & 31;
  int n0 = wv * 16;
  int mh = lane & 15, kh = (lane >> 4) << 1;
  const float* A = yg + (size_t)(m0 + mh) * 64 + kh;
  const float* Wp = Wo + (size_t)(n0 + mh) * 64 + kh;
  v8f acc = {};
#pragma unroll
  for (int k = 0; k < 64; k += 4) {
    v2f a = {A[k], A[k + 1]};
    v2f b = {Wp[k], Wp[k + 1]};
    acc = wmma4(a, b, acc);
  }
  int n = n0 + (lane & 15);
  int mb = (lane >> 4) * 8;
#pragma unroll
  for (int v = 0; v < 8; ++v) {
    int m = m0 + mb + v;
    if (mode == 0) {
      vim[(size_t)m * 32 + n] = acc[v] + vb[n] + tok[(size_t)m * 32 + n];
    } else {
      int l = m & (LTOK - 1);
      int dm = (m - l) + (LTOK - 1 - l);
      vim[(size_t)dm * 32 + n] += acc[v];
    }
  }
}

// ---------------------------------------------------------------------------
// 8) aux: depthwise 3x3 (ch 0-15) / 5x5 (ch 16-31) + bias; BN stats
// ---------------------------------------------------------------------------
__global__ void mk_kernel(const float* __restrict__ auxp,
                          const float* __restrict__ w3, const float* __restrict__ b3,
                          const float* __restrict__ w5, const float* __restrict__ b5,
                          float* __restrict__ mkb, float* __restrict__ stats) {
  int idx = blockIdx.x * 256 + threadIdx.x;
  int f = idx & 63;
  int t = (idx >> 6) & 255;
  int c = (idx >> 14) & 31;  // uniform per block
  int b = idx >> 19;
  const float* pl = auxp + ((size_t)(b * 32 + c)) * LTOK;
  float acc;
  if (c < 16) {
    acc = b3[c];
#pragma unroll
    for (int di = -1; di <= 1; ++di)
#pragma unroll
      for (int dj = -1; dj <= 1; ++dj) {
        int tt = t + di, ff = f + dj;
        if (tt >= 0 && tt < 256 && ff >= 0 && ff < 64)
          acc += w3[c * 9 + (di + 1) * 3 + (dj + 1)] * pl[tt * 64 + ff];
      }
  } else {
    int c5 = c - 16;
    acc = b5[c5];
    for (int di = -2; di <= 2; ++di)
      for (int dj = -2; dj <= 2; ++dj) {
        int tt = t + di, ff = f + dj;
        if (tt >= 0 && tt < 256 && ff >= 0 && ff < 64)
          acc += w5[c5 * 25 + (di + 2) * 5 + (dj + 2)] * pl[tt * 64 + ff];
      }
  }
  mkb[((size_t)(b * 32 + c)) * LTOK + t * 64 + f] = acc;
  float s1 = wredsum(acc);
  float s2 = wredsum(acc * acc);
  __shared__ float sm[16];
  int lane = threadIdx.x & 31;
  int wv = threadIdx.x >> 5;
  if (lane == 0) { sm[wv] = s1; sm[8 + wv] = s2; }
  __syncthreads();
  if (threadIdx.x == 0) {
    float a1 = 0.f;
    for (int i = 0; i < 8; ++i) a1 += sm[i];
    atomAddF(&stats[64 + c], a1);
  }
  if (threadIdx.x == 32) {
    float a2 = 0.f;
    for (int i = 0; i < 8; ++i) a2 += sm[8 + i];
    atomAddF(&stats[192 + 64 + c], a2);
  }
}

// ---------------------------------------------------------------------------
// 9) fuse 1x1 WMMA GEMM over concat(main_out, aux_out): M=NTOK N=64 K=64
// ---------------------------------------------------------------------------
__global__ void gemm_fuse_kernel(const float* __restrict__ vim, const float* __restrict__ mkb,
                                 const float* __restrict__ fw, const float* __restrict__ fb,
                                 float* __restrict__ fused, float* __restrict__ stats) {
  int m0 = blockIdx.x * 16;
  int wv = threadIdx.x >> 5;  // 4 waves
  int lane = threadIdx.x & 31;
  int n0 = wv * 16;
  int mh = lane & 15, kh = (lane >> 4) << 1;
  int m = m0 + mh;
  int bt = m >> 14, l = m & (LTOK - 1);
  const float* Wp = fw + (size_t)(n0 + mh) * 64 + kh;
  v8f acc = {};
#pragma unroll
  for (int k = 0; k < 64; k += 4) {
    int kk = k + kh;
    v2f a;
    if (k < 32) {
      a.x = vim[(size_t)m * 32 + kk];
      a.y = vim[(size_t)m * 32 + kk + 1];
    } else {
      int c = kk - 32;
      a.x = mkb[((size_t)(bt * 32 + c)) * LTOK + l];
      a.y = mkb[((size_t)(bt * 32 + c + 1)) * LTOK + l];
    }
    v2f b = {Wp[k], Wp[k + 1]};
    acc = wmma4(a, b, acc);
  }
  int n = n0 + (lane & 15);
  float fbn = fb[n];
  int mb = (lane >> 4) * 8;
  float s1 = 0.f, s2 = 0.f;
#pragma unroll
  for (int v = 0; v < 8; ++v) {
    float val = acc[v] + fbn;
    fused[(size_t)(m0 + mb + v) * 64 + n] = val;
    s1 += val;
    s2 += val * val;
  }
  s1 += __shfl_xor(s1, 16);
  s2 += __shfl_xor(s2, 16);
  if (lane < 16) { atomAddF(&stats[96 + n], s1); atomAddF(&stats[192 + 96 + n], s2); }
}

// ---------------------------------------------------------------------------
// 10) reconstruction WMMA GEMM (ConvTranspose 2x2 == pixel shuffle):
//     A = BN+ReLU(fused) on the fly; B[k][n] = rec_w[k*128+n]; N=128 (c,i,j)
// ---------------------------------------------------------------------------
__global__ void gemm_rec_kernel(const float* __restrict__ fused, const float* __restrict__ bnp,
                                const float* __restrict__ recw, const float* __restrict__ recb,
                                float* __restrict__ out, float* __restrict__ stats) {
  __shared__ float sc[64], sh[64];
  if (threadIdx.x < 64) {
    sc[threadIdx.x] = bnp[2 * (96 + threadIdx.x)];
    sh[threadIdx.x] = bnp[2 * (96 + threadIdx.x) + 1];
  }
  __syncthreads();
  int m0 = blockIdx.x * 16;
  int wv = threadIdx.x >> 5;  // 8 waves
  int lane = threadIdx.x & 31;
  int n0 = wv * 16;
  int mh = lane & 15, kh = (lane >> 4) << 1;
  const float* A = fused + (size_t)(m0 + mh) * 64;
  int nB = n0 + mh;
  v8f acc = {};
#pragma unroll
  for (int k = 0; k < 64; k += 4) {
    int kk = k + kh;
    v2f a;
    a.x = fmaxf(A[kk] * sc[kk] + sh[kk], 0.f);
    a.y = fmaxf(A[kk + 1] * sc[kk + 1] + sh[kk + 1], 0.f);
    v2f b;
    b.x = recw[(size_t)kk * 128 + nB];
    b.y = recw[(size_t)(kk + 1) * 128 + nB];
    acc = wmma4(a, b, acc);
  }
  int n = n0 + (lane & 15);
  int c = n >> 2, ii = (n >> 1) & 1, jj = n & 1;
  float rb = recb[c];
  int mb = (lane >> 4) * 8;
  float s1 = 0.f, s2 = 0.f;
#pragma unroll
  for (int v = 0; v < 8; ++v) {
    int m = m0 + mb + v;
    int bt = m >> 14;
    int l = m & (LTOK - 1);
    int t = l >> 6, f = l & 63;
    float val = acc[v] + rb;
    out[(((size_t)(bt * 32 + c) * 512) + (2 * t + ii)) * 128 + (2 * f + jj)] = val;
    s1 += val;
    s2 += val * val;
  }
  s1 += __shfl_xor(s1, 16);
  s2 += __shfl_xor(s2, 16);
  if (lane < 16) { atomAddF(&stats[160 + c], s1); atomAddF(&stats[192 + 160 + c], s2); }
}

// ---------------------------------------------------------------------------
// 11) final: BN+ReLU on reconstruction + residual, in place on d_out
// ---------------------------------------------------------------------------
__global__ void final_kernel(float* __restrict__ out, const float* __restrict__ x,
                             const float* __restrict__ bnp) {
  int idx = blockIdx.x * 256 + threadIdx.x;
  int c = (idx >> 16) & 31;  // plane = 512*128
  float v = out[idx] * bnp[2 * (160 + c)] + bnp[2 * (160 + c) + 1];
  out[idx] = fmaxf(v, 0.f) + x[idx];
}

// ---------------------------------------------------------------------------
extern "C" void kernel_launch(void* const* d_in, const int* in_sizes, int n_in,
                              void* d_out, int out_size, void* d_ws, size_t ws_size,
                              hipStream_t stream) {
  const float* X = (const float*)d_in[0];
  const float* main_w = (const float*)d_in[1];
  const float* main_b = (const float*)d_in[2];
  const float* main_bng = (const float*)d_in[3];
  const float* main_bnb = (const float*)d_in[4];
  const float* aux_w = (const float*)d_in[5];
  const float* aux_b = (const float*)d_in[6];
  const float* aux_bng = (const float*)d_in[7];
  const float* aux_bnb = (const float*)d_in[8];
  const float* ln_g = (const float*)d_in[9];
  const float* ln_b = (const float*)d_in[10];
  const float* in_w[2] = {(const float*)d_in[11], (const float*)d_in[20]};
  const float* conv_w[2] = {(const float*)d_in[12], (const float*)d_in[21]};
  const float* conv_b[2] = {(const float*)d_in[13], (const float*)d_in[22]};
  const float* xproj_w[2] = {(const float*)d_in[14], (const float*)d_in[23]};
  const float* dt_w[2] = {(const float*)d_in[15], (const float*)d_in[24]};
  const float* dt_b[2] = {(const float*)d_in[16], (const float*)d_in[25]};
  const float* A_log[2] = {(const float*)d_in[17], (const float*)d_in[26]};
  const float* Dp[2] = {(const float*)d_in[18], (const float*)d_in[27]};
  const float* out_w[2] = {(const float*)d_in[19], (const float*)d_in[28]};
  const float* vim_out_w = (const float*)d_in[29];
  const float* vim_out_b = (const float*)d_in[30];
  const float* mk3_w = (const float*)d_in[31];
  const float* mk3_b = (const float*)d_in[32];
  const float* mk5_w = (const float*)d_in[33];
  const float* mk5_b = (const float*)d_in[34];
  const float* mk_bng = (const float*)d_in[35];
  const float* mk_bnb = (const float*)d_in[36];
  const float* fuse_w = (const float*)d_in[37];
  const float* fuse_b = (const float*)d_in[38];
  const float* fuse_bng = (const float*)d_in[39];
  const float* fuse_bnb = (const float*)d_in[40];
  const float* rec_w = (const float*)d_in[41];
  const float* rec_b = (const float*)d_in[42];
  const float* rec_bng = (const float*)d_in[43];
  const float* rec_bnb = (const float*)d_in[44];
  float* out = (float*)d_out;

  float* W = (float*)d_ws;
  size_t o = 0;
  auto alloc = [&](size_t n) {
    float* p = W + o;
    o += (n + 255) & ~(size_t)255;
    return p;
  };
  float* mainp = alloc(4194304);   // [B,32,L]
  float* auxp = alloc(4194304);
  float* tok = alloc(4194304);     // [NTOK,32]
  float* xn = alloc(4194304);
  float* u_raw = alloc(8388608);   // [NTOK,64]; reused as yg after dwconv
  float* u_act = alloc(8388608);
  float* zb = alloc(8388608);
  float* dbc2 = alloc(12582912);   // [NTOK,96]
  float* vim = alloc(4194304);
  float* mkb = alloc(4194304);
  float* fused = alloc(8388608);   // [NTOK,64]
  float* hend = alloc(2097152);    // [B*NCH, 64,16] chunk-final states
  float* hin = alloc(2097152);     // [B*NCH, 64,16] chunk incoming states
  float* sdt = alloc(131072);      // [B*NCH, 64] chunk sum(dt)
  float* Aneg[2] = {alloc(1024), alloc(1024)};
  float* Wc[2] = {alloc(6144), alloc(6144)};
  float* Wo[2] = {alloc(2048), alloc(2048)};
  float* stats = alloc(384);       // sums[0..191], sumsq[192..383]
  float* bnp = alloc(384);         // (scale, shift) per channel
  float* yg = u_raw;

  prep_kernel<<<1, 256, 0, stream>>>(A_log[0], A_log[1], dt_w[0], dt_w[1], xproj_w[0],
                                     xproj_w[1], vim_out_w, out_w[0], out_w[1], Aneg[0],
                                     Aneg[1], Wc[0], Wc[1], Wo[0], Wo[1], stats);
  head_kernel<<<512, 256, 0, stream>>>(X, main_w, main_b, aux_w, aux_b, mainp, auxp, stats);
  bn_finalize_kernel<<<1, 64, 0, stream>>>(stats, main_bng, main_bnb, bnp, 0, 32, 1.f / 131072.f);
  bn_finalize_kernel<<<1, 64, 0, stream>>>(stats, aux_bng, aux_bnb, bnp, 32, 32, 1.f / 131072.f);
  bn_act_kernel<<<16384, 256, 0, stream>>>(mainp, bnp, 0, 0);
  bn_act_kernel<<<16384, 256, 0, stream>>>(auxp, bnp, 32, 0);
  ln_kernel<<<512, 256, 0, stream>>>(mainp, ln_g, ln_b, tok, xn);

  for (int dir = 0; dir < 2; ++dir) {  // 0 = fwd, 1 = bwd (flipped)
    gemm_inproj_kernel<<<8192, 256, 0, stream>>>(xn, in_w[dir], u_raw, zb, dir);
    dwconv_silu_kernel<<<32768, 256, 0, stream>>>(u_raw, conv_w[dir], conv_b[dir], u_act);
    gemm_xproj_kernel<<<8192, 192, 0, stream>>>(u_act, Wc[dir], dbc2);
    scan_phase1_kernel<<<8 * NCH, 256, 0, stream>>>(dbc2, u_act, Aneg[dir], dt_b[dir], hend,
                                                    sdt);
    scan_phase2_kernel<<<8, 256, 0, stream>>>(hend, sdt, Aneg[dir], hin);
    scan_phase3_kernel<<<8 * NCH, 256, 0, stream>>>(dbc2, u_act, zb, Aneg[dir], dt_b[dir],
                                                    Dp[dir], hin, yg);
    gemm_out_kernel<<<8192, 64, 0, stream>>>(yg, Wo[dir], vim_out_b, tok, vim, dir);
  }

  mk_kernel<<<16384, 256, 0, stream>>>(auxp, mk3_w, mk3_b, mk5_w, mk5_b, mkb, stats);
  bn_finalize_kernel<<<1, 64, 0, stream>>>(stats, mk_bng, mk_bnb, bnp, 64, 32, 1.f / 131072.f);
  bn_act_kernel<<<16384, 256, 0, stream>>>(mkb, bnp, 64, 1);
  gemm_fuse_kernel<<<8192, 128, 0, stream>>>(vim, mkb, fuse_w, fuse_b, fused, stats);
  bn_finalize_kernel<<<1, 64, 0, stream>>>(stats, fuse_bng, fuse_bnb, bnp, 96, 64, 1.f / 131072.f);
  gemm_rec_kernel<<<8192, 256, 0, stream>>>(fused, bnp, rec_w, rec_b, out, stats);
  bn_finalize_kernel<<<1, 64, 0, stream>>>(stats, rec_bng, rec_bnb, bnp, 160, 32, 1.f / 524288.f);
  final_kernel<<<65536, 256, 0, stream>>>(out, X, bnp);

  (void)in_sizes; (void)n_in; (void)out_size; (void)ws_size;
}